// GNNModule_9869834846977
// MI455X (gfx1250) — compile-verified
//
#include <hip/hip_runtime.h>
#include <hip/hip_bf16.h>

typedef __attribute__((ext_vector_type(2))) float v2f;
typedef __attribute__((ext_vector_type(8))) float v8f;

#define D 64
#define NGRAPH 256
#define EPS 1e-5f

// ---------------------------------------------------------------- utilities
__global__ void zero_f32(float* __restrict__ p, int n) {
    int i = blockIdx.x * blockDim.x + threadIdx.x;
    if (i < n) p[i] = 0.0f;
}

// out_deg[src[e]] += 1 ; in_deg[dst[e]] += 1
__global__ void degree_kernel(const int* __restrict__ src,
                              const int* __restrict__ dst,
                              float* __restrict__ deg_out,
                              float* __restrict__ deg_in, int E) {
    int e = blockIdx.x * blockDim.x + threadIdx.x;
    if (e >= E) return;
    atomicAdd(&deg_out[src[e]], 1.0f);
    atomicAdd(&deg_in[dst[e]], 1.0f);
}

// rs = rsqrt(max(deg, 1))
__global__ void rsqrt_deg_kernel(const float* __restrict__ deg,
                                 float* __restrict__ rs, int n) {
    int i = blockIdx.x * blockDim.x + threadIdx.x;
    if (i < n) rs[i] = rsqrtf(fmaxf(deg[i], 1.0f));
}

// coef[e] = ew[e] * rsqrt(out_deg[src[e]])  — shared by both layers
__global__ void edge_coef_kernel(const float* __restrict__ ew,
                                 const int* __restrict__ src,
                                 const float* __restrict__ rs_out,
                                 float* __restrict__ coef, int E) {
    int e = blockIdx.x * blockDim.x + threadIdx.x;
    if (e < E) coef[e] = ew[e] * rs_out[src[e]];
}

// WT[n][k] = W[k][n]  (64x64)
__global__ void transpose_w_kernel(const float* __restrict__ W,
                                   float* __restrict__ WT) {
    int i = blockIdx.x * blockDim.x + threadIdx.x;   // i = k*64 + n
    if (i < D * D) WT[(i & 63) * D + (i >> 6)] = W[i];
}

// agg[dst[e]] += h[src[e]] * coef[e]   (16 float4-quads per 64-wide row)
// src-degree norm is pre-folded into coef, so we gather h directly.
__global__ void scatter_edges_kernel(const float* __restrict__ h,
                                     const int* __restrict__ src,
                                     const int* __restrict__ dst,
                                     const float* __restrict__ coef,
                                     float* __restrict__ agg, int E) {
    int gid = blockIdx.x * blockDim.x + threadIdx.x;
    int e = gid >> 4;
    int q = gid & 15;
    if (e >= E) return;
    int s = src[e], d = dst[e];
    float w = coef[e];
    const float4* row = (const float4*)(h + (size_t)s * D);
    float4 v = row[q];
    float* out = agg + (size_t)d * D + q * 4;
    atomicAdd(out + 0, v.x * w);
    atomicAdd(out + 1, v.y * w);
    atomicAdd(out + 2, v.z * w);
    atomicAdd(out + 3, v.w * w);
}

// Fused: y = relu( BN( relu( (agg * rsqrt(in_deg)) @ W + b ) ) )
// One wave per 16-node tile; 4 WMMA accumulators cover the full 64-wide output.
// V_WMMA_F32_16X16X4_F32: A 16x4, B 4x16, C/D 16x16 (8 VGPRs).
// WT is W transposed so each B fragment is one contiguous 64-bit load.
__global__ void conv_gemm_bn_relu_kernel(const float* __restrict__ agg,
                                         const float* __restrict__ rs_in,
                                         const float* __restrict__ WT,
                                         const float* __restrict__ bias,
                                         const float* __restrict__ gamma,
                                         const float* __restrict__ beta,
                                         const float* __restrict__ rm,
                                         const float* __restrict__ rv,
                                         float* __restrict__ out,
                                         int nNodes, int nTiles) {
    const int lane = threadIdx.x & 31;
    const int wid  = threadIdx.x >> 5;
    const int tile = blockIdx.x * (blockDim.x >> 5) + wid;
    if (tile >= nTiles) return;          // wave-uniform: EXEC stays all-ones

    const int mrow = tile * 16 + (lane & 15);
    const int r    = mrow < nNodes ? mrow : nNodes - 1;   // clamp loads only
    const int koff = (lane >> 4) * 2;    // lanes 16-31 hold K+2/K+3
    const int nsub = lane & 15;
    const float scale = rs_in[r];
    const float* arow = agg + (size_t)r * D;
    // per-lane transposed-W rows for the 4 output tiles
    const v2f* wt0 = (const v2f*)(WT + (size_t)(nsub +  0) * D);
    const v2f* wt1 = (const v2f*)(WT + (size_t)(nsub + 16) * D);
    const v2f* wt2 = (const v2f*)(WT + (size_t)(nsub + 32) * D);
    const v2f* wt3 = (const v2f*)(WT + (size_t)(nsub + 48) * D);

    v8f acc[4] = {v8f{}, v8f{}, v8f{}, v8f{}};

    #pragma unroll
    for (int k0 = 0; k0 < D; k0 += 4) {
        const int kk = k0 + koff;        // even: k0 % 4 == 0, koff in {0,2}
        // A fragment: agg[m][kk], agg[m][kk+1], pre-scaled by rsqrt(in_deg)
        v2f a = *(const v2f*)(arow + kk);
        a *= scale;
        // B fragments: {W[kk][n], W[kk+1][n]} = contiguous pair of WT[n]
        v2f b0 = wt0[kk >> 1];
        v2f b1 = wt1[kk >> 1];
        v2f b2 = wt2[kk >> 1];
        v2f b3 = wt3[kk >> 1];
        acc[0] = __builtin_amdgcn_wmma_f32_16x16x4_f32(false, a, false, b0,
                                                       (short)0, acc[0], false, false);
        acc[1] = __builtin_amdgcn_wmma_f32_16x16x4_f32(false, a, false, b1,
                                                       (short)0, acc[1], false, false);
        acc[2] = __builtin_amdgcn_wmma_f32_16x16x4_f32(false, a, false, b2,
                                                       (short)0, acc[2], false, false);
        acc[3] = __builtin_amdgcn_wmma_f32_16x16x4_f32(false, a, false, b3,
                                                       (short)0, acc[3], false, false);
    }

    // Epilogue: C layout — VGPR v, lanes 0-15: M=v, lanes 16-31: M=v+8; N=lane%16
    const int m_lo = (lane >> 4) * 8;
    #pragma unroll
    for (int t = 0; t < 4; ++t) {
        const int n = t * 16 + nsub;
        const float bn_s = gamma[n] * rsqrtf(rv[n] + EPS);
        const float bn_b = beta[n] - rm[n] * bn_s;
        const float bv   = bias[n];
        #pragma unroll
        for (int v = 0; v < 8; ++v) {
            const int row = tile * 16 + v + m_lo;
            float x = acc[t][v] + bv;
            x = fmaxf(x, 0.0f);          // GraphConv activation
            x = x * bn_s + bn_b;         // BatchNorm (eval)
            x = fmaxf(x, 0.0f);          // post-BN ReLU
            if (row < nNodes) out[(size_t)row * D + n] = x;
        }
    }
}

// per-graph sum + count (atomic), then mean
__global__ void readout_accum_kernel(const float* __restrict__ h2,
                                     const int* __restrict__ n2g,
                                     float* __restrict__ g_sum,
                                     float* __restrict__ g_cnt, int N) {
    int gid = blockIdx.x * blockDim.x + threadIdx.x;
    int n = gid >> 6, d = gid & 63;
    if (n >= N) return;
    int g = n2g[n];
    atomicAdd(&g_sum[(size_t)g * D + d], h2[(size_t)n * D + d]);
    if (d == 0) atomicAdd(&g_cnt[g], 1.0f);
}

__global__ void readout_div_kernel(float* __restrict__ g_sum,
                                   const float* __restrict__ g_cnt) {
    int g = blockIdx.x, d = threadIdx.x;
    float c = fmaxf(g_cnt[g], 1.0f);
    g_sum[(size_t)g * D + d] /= c;
}

// ---------------------------------------------------------------- launcher
extern "C" void kernel_launch(void* const* d_in, const int* in_sizes, int n_in,
                              void* d_out, int out_size, void* d_ws, size_t ws_size,
                              hipStream_t stream) {
    const float* h      = (const float*)d_in[0];
    const float* ew     = (const float*)d_in[1];
    const int*   src    = (const int*)  d_in[2];
    const int*   dst    = (const int*)  d_in[3];
    const int*   n2g    = (const int*)  d_in[4];
    const float* W1     = (const float*)d_in[5];
    const float* b1     = (const float*)d_in[6];
    const float* gamma1 = (const float*)d_in[7];
    const float* beta1  = (const float*)d_in[8];
    const float* rm1    = (const float*)d_in[9];
    const float* rv1    = (const float*)d_in[10];
    const float* W2     = (const float*)d_in[11];
    const float* b2     = (const float*)d_in[12];
    const float* gamma2 = (const float*)d_in[13];
    const float* beta2  = (const float*)d_in[14];
    const float* rm2    = (const float*)d_in[15];
    const float* rv2    = (const float*)d_in[16];
    float* out = (float*)d_out;

    const int N = in_sizes[0] / D;
    const int E = in_sizes[1];

    // workspace layout (floats)
    float* ws      = (float*)d_ws;
    float* deg_out = ws;                          // N
    float* deg_in  = deg_out + N;                 // N
    float* rs_out  = deg_in  + N;                 // N
    float* rs_in   = rs_out  + N;                 // N
    float* coef    = rs_in   + N;                 // E
    float* WT1     = coef + E;                    // D*D
    float* WT2     = WT1  + D * D;                // D*D
    float* agg     = WT2  + D * D;                // N*D
    float* h1      = agg + (size_t)N * D;         // N*D
    float* h2      = h1  + (size_t)N * D;         // N*D
    float* g_cnt   = h2  + (size_t)N * D;         // NGRAPH

    const int B = 256;
    const int nElem  = N * D;
    const int nTiles = (N + 15) / 16;
    const int gemmBlocks = (nTiles + 7) / 8;      // 8 waves / block

    // degrees + per-edge coefficient (shared by both layers)
    zero_f32<<<(2 * N + B - 1) / B, B, 0, stream>>>(deg_out, 2 * N);
    degree_kernel<<<(E + B - 1) / B, B, 0, stream>>>(src, dst, deg_out, deg_in, E);
    rsqrt_deg_kernel<<<(N + B - 1) / B, B, 0, stream>>>(deg_out, rs_out, N);
    rsqrt_deg_kernel<<<(N + B - 1) / B, B, 0, stream>>>(deg_in,  rs_in,  N);
    edge_coef_kernel<<<(E + B - 1) / B, B, 0, stream>>>(ew, src, rs_out, coef, E);
    transpose_w_kernel<<<(D * D + B - 1) / B, B, 0, stream>>>(W1, WT1);
    transpose_w_kernel<<<(D * D + B - 1) / B, B, 0, stream>>>(W2, WT2);

    // ---- layer 1
    zero_f32<<<(nElem + B - 1) / B, B, 0, stream>>>(agg, nElem);
    scatter_edges_kernel<<<(E * 16 + B - 1) / B, B, 0, stream>>>(h, src, dst, coef, agg, E);
    conv_gemm_bn_relu_kernel<<<gemmBlocks, B, 0, stream>>>(
        agg, rs_in, WT1, b1, gamma1, beta1, rm1, rv1, h1, N, nTiles);

    // ---- layer 2
    zero_f32<<<(nElem + B - 1) / B, B, 0, stream>>>(agg, nElem);
    scatter_edges_kernel<<<(E * 16 + B - 1) / B, B, 0, stream>>>(h1, src, dst, coef, agg, E);
    conv_gemm_bn_relu_kernel<<<gemmBlocks, B, 0, stream>>>(
        agg, rs_in, WT2, b2, gamma2, beta2, rm2, rv2, h2, N, nTiles);

    // ---- mean-nodes readout
    zero_f32<<<(NGRAPH * D + B - 1) / B, B, 0, stream>>>(out, NGRAPH * D);
    zero_f32<<<(NGRAPH + B - 1) / B, B, 0, stream>>>(g_cnt, NGRAPH);
    readout_accum_kernel<<<(nElem + B - 1) / B, B, 0, stream>>>(h2, n2g, out, g_cnt, N);
    readout_div_kernel<<<NGRAPH, D, 0, stream>>>(out, g_cnt);
}